// projector_68539088109902
// MI455X (gfx1250) — compile-verified
//
#include <hip/hip_runtime.h>
#include <hip/hip_bf16.h>

// Fanbeam forward projection, CDNA5 (gfx1250, wave32).
// One wave handles 16 rays x (1024/SPLIT) samples; V_WMMA_F32_16X16X4_F32
// against an all-ones B matrix performs the per-ray sample accumulation
// (row sums of the A tile land in the C accumulator), so the wave needs no
// cross-lane shuffles and no scalar accumulation chain. SPLIT=4 quadruples
// wave count for occupancy; a deterministic fixed-order reduction combines
// the partials (no float atomics -> bit-stable across graph replays).

typedef __attribute__((ext_vector_type(2))) float v2f;
typedef __attribute__((ext_vector_type(8))) float v8f;

#define N_IMG 512
#define N_ANGLES 25
#define N_DET 512
#define DET_SPACING 3.0f
#define SRC_DIST 512.0f
#define DET_DIST 512.0f
#define N_SAMPLES 1024
#define N_RAYS (N_ANGLES * N_DET)   // 12800

__device__ __forceinline__ float tex2d_zero(const float* __restrict__ img, int r, int c) {
    // zero outside the image, clamped address so the load itself is unconditional
    bool valid = ((unsigned)r < (unsigned)N_IMG) && ((unsigned)c < (unsigned)N_IMG);
    int rc = min(max(r, 0), N_IMG - 1);
    int cc = min(max(c, 0), N_IMG - 1);
    float v = img[rc * N_IMG + cc];
    return valid ? v : 0.0f;
}

__device__ __forceinline__ float bilin(const float* __restrict__ img, float row, float col) {
    float r0f = floorf(row), c0f = floorf(col);
    float fr = row - r0f, fc = col - c0f;
    int r0 = (int)r0f, c0 = (int)c0f;
    float v00 = tex2d_zero(img, r0, c0);
    float v01 = tex2d_zero(img, r0, c0 + 1);
    float v10 = tex2d_zero(img, r0 + 1, c0);
    float v11 = tex2d_zero(img, r0 + 1, c0 + 1);
    float top = fmaf(fc, v01 - v00, v00);
    float bot = fmaf(fc, v11 - v10, v10);
    return fmaf(fr, bot - top, top);
}

template <int SPLIT>
__global__ __launch_bounds__(32) void fanbeam_wmma_kernel(const float* __restrict__ img,
                                                          float* __restrict__ dst) {
    const int lane    = threadIdx.x;                  // wave32: 0..31
    const int grp     = (int)blockIdx.x / SPLIT;      // 16-ray group
    const int chunk   = (int)blockIdx.x % SPLIT;      // sample-range chunk
    const int rayBase = grp * 16;
    const int m       = lane & 15;                    // A-matrix row M for this lane
    const int kOff    = (lane < 16) ? 0 : 2;          // 16x4 f32 A layout: lanes 0-15 K=0,1; 16-31 K=2,3

    // Warm WGP$/L2 with the (L2-resident, 1MB) image; emits global_prefetch_b8.
    __builtin_prefetch(img + (size_t)lane * 8192, 0, 0);

    const int ray  = rayBase + m;
    const int aIdx = ray >> 9;                        // ray / N_DET, uniform across the wave
    const int det  = ray & (N_DET - 1);

    const float beta = (float)aIdx * (6.28318530717958647692f / (float)N_ANGLES);
    const float cb = cosf(beta), sb = sinf(beta);
    const float t    = ((float)det - (float)(N_DET - 1) * 0.5f) * DET_SPACING;
    const float srcX = -SRC_DIST * sb;
    const float srcY =  SRC_DIST * cb;
    const float dX   = (t * cb + DET_DIST * sb) - srcX;
    const float dY   = (t * sb - DET_DIST * cb) - srcY;
    const float segLen = sqrtf(dX * dX + dY * dY);
    const float step   = segLen * (1.0f / (float)N_SAMPLES); // per-ray step length
    const float half   = (float)(N_IMG - 1) * 0.5f;

    // Two independent WMMA accumulators break the D->C dependency chain and
    // keep 16 scattered gathers in flight per unrolled iteration.
    v8f acc0 = {}, acc1 = {};
    v2f ones;
    ones.x = 1.0f;
    ones.y = 1.0f;            // B = 4x16 all-ones -> WMMA computes A row sums into every C column

    const int itersTotal = N_SAMPLES / 4 / SPLIT;     // 4-sample chunks handled by this wave
    const int itBeg      = chunk * itersTotal;

    for (int it2 = 0; it2 < itersTotal / 2; ++it2) {
        const int itA = itBeg + it2 * 2;

        {   // ---- sample pair A -> acc0 ----
            const int s0 = itA * 4 + kOff;
            const float ts0 = ((float)s0 + 0.5f) * (1.0f / (float)N_SAMPLES);
            const float ts1 = ((float)s0 + 1.5f) * (1.0f / (float)N_SAMPLES);
            const float x0 = fmaf(ts0, dX, srcX), y0 = fmaf(ts0, dY, srcY);
            const float x1 = fmaf(ts1, dX, srcX), y1 = fmaf(ts1, dY, srcY);
            v2f a;
            a.x = bilin(img, half - y0, x0 + half) * step;
            a.y = bilin(img, half - y1, x1 + half) * step;
            acc0 = __builtin_amdgcn_wmma_f32_16x16x4_f32(false, a, false, ones,
                                                         (short)0, acc0, false, false);
        }
        {   // ---- sample pair B -> acc1 (independent chain) ----
            const int s0 = (itA + 1) * 4 + kOff;
            const float ts0 = ((float)s0 + 0.5f) * (1.0f / (float)N_SAMPLES);
            const float ts1 = ((float)s0 + 1.5f) * (1.0f / (float)N_SAMPLES);
            const float x0 = fmaf(ts0, dX, srcX), y0 = fmaf(ts0, dY, srcY);
            const float x1 = fmaf(ts1, dX, srcX), y1 = fmaf(ts1, dY, srcY);
            v2f a;
            a.x = bilin(img, half - y0, x0 + half) * step;
            a.y = bilin(img, half - y1, x1 + half) * step;
            acc1 = __builtin_amdgcn_wmma_f32_16x16x4_f32(false, a, false, ones,
                                                         (short)0, acc1, false, false);
        }
    }

    v8f acc;
#pragma unroll
    for (int r = 0; r < 8; ++r) acc[r] = acc0[r] + acc1[r];

    // C/D 16x16 f32 layout: lanes 0-15 hold M=0..7 in VGPR0..7 (col = lane, all
    // cols equal); lanes 16-31 hold M=8..15. Lane 0 writes rays 0..7, lane 16
    // writes rays 8..15, into this chunk's partial slice.
    float* p = dst + (size_t)chunk * N_RAYS;
    if (lane == 0) {
#pragma unroll
        for (int r = 0; r < 8; ++r) p[rayBase + r] = acc[r];
    } else if (lane == 16) {
#pragma unroll
        for (int r = 0; r < 8; ++r) p[rayBase + 8 + r] = acc[r];
    }
}

__global__ __launch_bounds__(256) void reduce4_kernel(const float* __restrict__ partial,
                                                      float* __restrict__ out) {
    int i = (int)blockIdx.x * 256 + (int)threadIdx.x;
    if (i < N_RAYS) {
        // fixed summation order -> deterministic output
        out[i] = ((partial[i] + partial[N_RAYS + i]) +
                  (partial[2 * N_RAYS + i] + partial[3 * N_RAYS + i]));
    }
}

extern "C" void kernel_launch(void* const* d_in, const int* in_sizes, int n_in,
                              void* d_out, int out_size, void* d_ws, size_t ws_size,
                              hipStream_t stream) {
    (void)in_sizes; (void)n_in; (void)out_size;
    const float* img = (const float*)d_in[0];   // [1,1,512,512] fp32
    float* out = (float*)d_out;                 // [1,1,25,512] fp32

    const size_t needWs = (size_t)4 * N_RAYS * sizeof(float);   // 200 KB of partials
    if (d_ws != nullptr && ws_size >= needWs) {
        // 4-way sample split: 3200 waves for occupancy, then deterministic reduce.
        float* partial = (float*)d_ws;
        fanbeam_wmma_kernel<4><<<(N_RAYS / 16) * 4, 32, 0, stream>>>(img, partial);
        reduce4_kernel<<<(N_RAYS + 255) / 256, 256, 0, stream>>>(partial, out);
    } else {
        // Fallback: single pass straight into the output.
        fanbeam_wmma_kernel<1><<<N_RAYS / 16, 32, 0, stream>>>(img, out);
    }
}